// ProposalLayer_4372276707983
// MI455X (gfx1250) — compile-verified
//
#include <hip/hip_runtime.h>
#include <hip/hip_bf16.h>
#include <stdint.h>

// ---------------- problem constants ----------------
#define BB        8
#define AA        262144
#define NMS_LIM   6000
#define PROP      2000
#define NBINS     4096
#define CAND      8192          // candidate capacity per image (>= NMS_LIM + max bin pop)
#define IOU_TH    0.5f

// ---------------- workspace layout (bytes) ----------------
#define HIST_OFF   0
#define CNT_OFF    (BB * NBINS * 4)                   // 131072
#define CUT_OFF    (CNT_OFF + BB * 4)                 // 131104
#define KEYS_OFF   (CUT_OFF + BB * 4 + 28)            // 131164, only 4B aligned
#define BOXES_OFF  (KEYS_OFF + BB * CAND * 8)         // 655452, only 4B aligned
#define SCORE_OFF  (BOXES_OFF + BB * NMS_LIM * 16)    // 1,423,452
// total = SCORE_OFF + BB*AA*4 = ~9.8 MB

typedef unsigned long long u64;

// ---------------- CDNA5 async global->LDS helpers (ASYNCcnt path) --------
__device__ __forceinline__ void async_g2l_b128(unsigned lds_addr, unsigned voff,
                                               const void* sbase) {
    asm volatile("global_load_async_to_lds_b128 %0, %1, %2"
                 :
                 : "v"(lds_addr), "v"(voff), "s"(sbase)
                 : "memory");
}
__device__ __forceinline__ void async_g2l_b64(unsigned lds_addr, unsigned voff,
                                              const void* sbase) {
    asm volatile("global_load_async_to_lds_b64 %0, %1, %2"
                 :
                 : "v"(lds_addr), "v"(voff), "s"(sbase)
                 : "memory");
}
__device__ __forceinline__ void wait_asynccnt0() {
#if __has_builtin(__builtin_amdgcn_s_wait_asynccnt)
    __builtin_amdgcn_s_wait_asynccnt(0);
#else
    asm volatile("s_wait_asynccnt 0" ::: "memory");
#endif
}

// ---------------- kernel 0: zero hist/cnt/cut and d_out ----------------
__global__ void k_zero(unsigned* __restrict__ meta, int nmeta,
                       float* __restrict__ out, int nout) {
    int g = blockIdx.x * blockDim.x + threadIdx.x;
    if (g < nmeta) meta[g] = 0u;
    if (g < nout)  out[g] = 0.0f;
}

// ---------------- kernel 1: fused score extract + histogram ----------------
// grid = BB * 64 blocks of 256 threads; each block covers 4096 anchors.
__global__ void k_score_hist(const float* __restrict__ probs,
                             unsigned* __restrict__ hist,
                             float* __restrict__ scores) {
    __shared__ unsigned lh[NBINS];
    const int tid = threadIdx.x;
    for (int i = tid; i < NBINS; i += 256) lh[i] = 0u;
    __syncthreads();

    const int b     = blockIdx.x >> 6;            // / 64
    const int chunk = blockIdx.x & 63;
    const size_t base = (size_t)b * AA + (size_t)chunk * 4096;
    const float2* p2 = (const float2*)probs;      // [B*A] pairs (neg, pos)
    #pragma unroll 4
    for (int k = 0; k < 16; ++k) {
        size_t a = base + tid + k * 256;
        float s = p2[a].y;                        // positive-class prob
        scores[a] = s;
        int bin = (int)(s * (float)NBINS);
        bin = bin < 0 ? 0 : (bin > NBINS - 1 ? NBINS - 1 : bin);
        atomicAdd(&lh[bin], 1u);
    }
    __syncthreads();
    unsigned* gh = hist + (size_t)b * NBINS;
    for (int i = tid; i < NBINS; i += 256)
        if (lh[i]) atomicAdd(&gh[i], lh[i]);
}

// ---------------- kernel 2: find cutoff bin per image ----------------
__global__ void k_cutoff(const unsigned* __restrict__ hist, unsigned* __restrict__ cut) {
    const int b = blockIdx.x;
    if (threadIdx.x == 0) {
        const unsigned* h = hist + (size_t)b * NBINS;
        unsigned cum = 0; int c = 0;
        for (int bin = NBINS - 1; bin >= 0; --bin) {
            cum += h[bin];
            if (cum >= (unsigned)NMS_LIM) { c = bin; break; }
        }
        cut[b] = (unsigned)c;
    }
}

// ---------------- kernel 3: compact candidates above cutoff ----------------
__global__ void k_compact(const float* __restrict__ scores,
                          const unsigned* __restrict__ cut,
                          unsigned* __restrict__ cnt,
                          u64* __restrict__ keys) {
    const int g = blockIdx.x * blockDim.x + threadIdx.x;   // 0 .. B*A-1
    const int b = g / AA;
    const int a = g - b * AA;
    float s = scores[g];
    int bin = (int)(s * (float)NBINS);
    bin = bin < 0 ? 0 : (bin > NBINS - 1 ? NBINS - 1 : bin);
    if ((unsigned)bin >= cut[b]) {
        unsigned pos = atomicAdd(&cnt[b], 1u);
        if (pos < CAND) {
            // descending score, ties -> ascending anchor index (== lax.top_k order)
            u64 key = ((u64)__float_as_uint(s) << 32) | (u64)(0xFFFFFFFFu - (unsigned)a);
            keys[(size_t)b * CAND + pos] = key;
        }
    }
}

// ---------------- kernel 4: async-stage keys, LDS bitonic sort, decode ----
// one block (1024 thr) per image; dynamic LDS = CAND*8 = 64 KB
__global__ void k_sort_decode(const u64* __restrict__ keysG,
                              const unsigned* __restrict__ cnt,
                              const float* __restrict__ rpn_bbox,
                              const float* __restrict__ anchors,
                              float* __restrict__ boxesWS) {
    extern __shared__ u64 key[];                  // [CAND]
    const int b = blockIdx.x, tid = threadIdx.x, NT = blockDim.x;
    unsigned n = cnt[b]; if (n > CAND) n = CAND;
    const u64* kg = keysG + (size_t)b * CAND;

    // valid entries: async global->LDS (b64); pad region: ds-store 0
    for (int i = tid; i < CAND; i += NT) {
        if (i < (int)n) {
            unsigned lds_addr = (unsigned)(uintptr_t)(&key[i]);
            async_g2l_b64(lds_addr, (unsigned)(i * 8), (const void*)kg);
        } else {
            key[i] = 0ull;                        // sorts to the tail (descending)
        }
    }
    wait_asynccnt0();
    __syncthreads();

    // bitonic, descending
    for (unsigned k = 2; k <= CAND; k <<= 1) {
        for (unsigned j = k >> 1; j > 0; j >>= 1) {
            for (unsigned i = tid; i < CAND; i += NT) {
                unsigned ixj = i ^ j;
                if (ixj > i) {
                    u64 x = key[i], y = key[ixj];
                    bool desc = ((i & k) == 0);
                    if (desc ? (x < y) : (x > y)) { key[i] = y; key[ixj] = x; }
                }
            }
            __syncthreads();
        }
    }

    // decode top NMS_LIM boxes (delta2box with RPN_BBOX_STD_DEV = .1,.1,.2,.2)
    float4* ob = (float4*)boxesWS + (size_t)b * NMS_LIM;
    for (int i = tid; i < NMS_LIM; i += NT) {
        unsigned idx = 0xFFFFFFFFu - (unsigned)(key[i] & 0xFFFFFFFFull);
        float4 box = make_float4(0.f, 0.f, 0.f, 0.f);
        if (idx < (unsigned)AA) {
            const float4 d  = ((const float4*)rpn_bbox)[(size_t)b * AA + idx];
            const float4 an = ((const float4*)anchors )[(size_t)b * AA + idx];
            float h  = an.z - an.x, w = an.w - an.y;
            float cy = an.x + 0.5f * h + d.x * 0.1f * h;
            float cx = an.y + 0.5f * w + d.y * 0.1f * w;
            float nh = h * __expf(d.z * 0.2f);
            float nw = w * __expf(d.w * 0.2f);
            box = make_float4(cy - 0.5f * nh, cx - 0.5f * nw,
                              cy + 0.5f * nh, cx + 0.5f * nw);
        }
        ob[i] = box;
    }
}

// ---------------- kernel 5: async-stage boxes, greedy NMS, pack -----------
// one block (1024 thr) per image; dynamic LDS ~123.2 KB (OK: 320 KB/WGP on CDNA5)
#define L_BOX   0
#define L_AREA  (NMS_LIM * 16)                    // 96000
#define L_SUP   (L_AREA + NMS_LIM * 4)            // 120000
#define L_RANK  (L_SUP + NMS_LIM)                 // 126000 (4B aligned)
#define L_TOTAL (L_RANK + 16)

__global__ void k_nms_pack(const float* __restrict__ boxesWS,
                           float* __restrict__ out) {
    extern __shared__ unsigned char smem[];
    float4*        lbox  = (float4*)(smem + L_BOX);
    float*         larea = (float*) (smem + L_AREA);
    unsigned char* sup   =          (smem + L_SUP);
    int*           lrank = (int*)   (smem + L_RANK);

    const int b = blockIdx.x, tid = threadIdx.x, NT = blockDim.x;
    const float4* gb = (const float4*)boxesWS + (size_t)b * NMS_LIM;

    // async global -> LDS stage of 6000 boxes (96 KB), ASYNCcnt tracked
    for (int i = tid; i < NMS_LIM; i += NT) {
        unsigned lds_addr = (unsigned)(uintptr_t)(&lbox[i]);
        async_g2l_b128(lds_addr, (unsigned)(i * 16), (const void*)gb);
    }
    wait_asynccnt0();
    __syncthreads();

    // areas, flags, zero this image's output slab
    for (int i = tid; i < NMS_LIM; i += NT) {
        float4 v = lbox[i];
        larea[i] = fmaxf(v.z - v.x, 0.f) * fmaxf(v.w - v.y, 0.f);
        sup[i] = 0;
    }
    if (tid == 0) *lrank = 0;
    float* ob = out + (size_t)b * PROP * 4;
    for (int i = tid; i < PROP * 4; i += NT) ob[i] = 0.f;
    __syncthreads();

    // greedy NMS, packing kept boxes in score order as we go
    float4* obox = (float4*)ob;
    for (int i = 0; i < NMS_LIM; ++i) {
        if (sup[i]) continue;          // uniform read; prior writes fenced by barrier
        float4 bi = lbox[i];
        float  ai = larea[i];
        for (int j = i + 1 + tid; j < NMS_LIM; j += NT) {
            if (!sup[j]) {
                float4 bj = lbox[j];
                float yy1 = fmaxf(bi.x, bj.x), xx1 = fmaxf(bi.y, bj.y);
                float yy2 = fminf(bi.z, bj.z), xx2 = fminf(bi.w, bj.w);
                float inter = fmaxf(yy2 - yy1, 0.f) * fmaxf(xx2 - xx1, 0.f);
                float uni = ai + larea[j] - inter;
                if (uni > 0.f && inter > IOU_TH * uni) sup[j] = 1;
            }
        }
        if (tid == 0) {
            int r = *lrank;
            if (r < PROP) obox[r] = bi;
            *lrank = r + 1;
        }
        __syncthreads();
    }
}

// ---------------- host-side launcher ----------------
extern "C" void kernel_launch(void* const* d_in, const int* in_sizes, int n_in,
                              void* d_out, int out_size, void* d_ws, size_t ws_size,
                              hipStream_t stream) {
    const float* probs    = (const float*)d_in[0];   // [B,A,2]
    const float* rpn_bbox = (const float*)d_in[1];   // [B,A,4]
    const float* anchors  = (const float*)d_in[2];   // [B,A,4]
    float* out = (float*)d_out;                      // [B,2000,4]

    unsigned char* ws = (unsigned char*)d_ws;
    unsigned* hist  = (unsigned*)(ws + HIST_OFF);
    unsigned* cnt   = (unsigned*)(ws + CNT_OFF);
    unsigned* cut   = (unsigned*)(ws + CUT_OFF);
    u64*      keys  = (u64*)     (ws + KEYS_OFF);
    float*  boxesWS = (float*)   (ws + BOXES_OFF);
    float*  scores  = (float*)   (ws + SCORE_OFF);

    const int nmeta = BB * NBINS + 2 * BB;           // hist+cnt+cut (contiguous u32)
    const int nout  = BB * PROP * 4;

    k_zero      <<<(nout + 255) / 256, 256, 0, stream>>>(hist, nmeta, out, nout);
    k_score_hist<<<BB * 64, 256, 0, stream>>>(probs, hist, scores);
    k_cutoff    <<<BB, 32, 0, stream>>>(hist, cut);
    k_compact   <<<(BB * AA) / 256, 256, 0, stream>>>(scores, cut, cnt, keys);
    k_sort_decode<<<BB, 1024, CAND * sizeof(u64), stream>>>(keys, cnt, rpn_bbox,
                                                            anchors, boxesWS);
    k_nms_pack   <<<BB, 1024, L_TOTAL, stream>>>(boxesWS, out);
}